// GINModel_67748814127260
// MI455X (gfx1250) — compile-verified
//
#include <hip/hip_runtime.h>
#include <hip/hip_bf16.h>

// GIN model for MI455X (gfx1250), fp32 end-to-end using V_WMMA_F32_16X16X4_F32.
// - One wave computes a full 16-row output strip (4 or 8 16x16 tiles) so the
//   fused A-fragment (combine / BN+ReLU) is loaded once per K-step.
// - Edge aggregation streams float4 gathers + f32 atomics through L2
//   (entire working set ~103MB < 192MB L2).
// - Wave-uniform full-tile store fast path avoids per-element exec juggling.

typedef float v2f __attribute__((ext_vector_type(2)));
typedef float v8f __attribute__((ext_vector_type(8)));

static __device__ __forceinline__ v8f wmma_f32_k4(v2f a, v2f b, v8f c) {
  // D = A(16x4) * B(4x16) + C(16x16), fp32
  return __builtin_amdgcn_wmma_f32_16x16x4_f32(
      /*neg_a=*/false, a, /*neg_b=*/false, b,
      /*c_mod=*/(short)0, c, /*reuse_a=*/false, /*reuse_b=*/false);
}

// ---------------------------------------------------------------------------
// Kernel 1: x_out[N,64] = x[N,32] @ Win[32,64] + bin
// One wave -> 16-row strip (4 col tiles), 8 K-steps of 4.
// ---------------------------------------------------------------------------
__global__ __launch_bounds__(256)
void gemm_in_kernel(const float* __restrict__ x, const float* __restrict__ Win,
                    const float* __restrict__ bin, float* __restrict__ xout,
                    int N, int mtiles) {
  const int lane  = threadIdx.x & 31;
  const int mtile = blockIdx.x * 8 + (threadIdx.x >> 5);
  if (mtile >= mtiles) return;           // wave-uniform; EXEC stays all-1s
  const int m0 = mtile * 16;
  const int half = lane >> 4, l16 = lane & 15;
  const int mrow = min(m0 + l16, N - 1); // branchless clamp keeps EXEC full
  v8f acc[4] = {};
#pragma unroll
  for (int kk = 0; kk < 32; kk += 4) {
    const int ka = kk + 2 * half;
    v2f a;
    a.x = x[(size_t)mrow * 32 + ka];
    a.y = x[(size_t)mrow * 32 + ka + 1];
#pragma unroll
    for (int nt = 0; nt < 4; ++nt) {
      v2f b;
      b.x = Win[(size_t)ka * 64 + nt * 16 + l16];
      b.y = Win[(size_t)(ka + 1) * 64 + nt * 16 + l16];
      acc[nt] = wmma_f32_k4(a, b, acc[nt]);
    }
  }
  if (m0 + 16 <= N) {                    // fast path: straight-line stores
#pragma unroll
    for (int nt = 0; nt < 4; ++nt) {
      const float bias = bin[nt * 16 + l16];
#pragma unroll
      for (int r = 0; r < 8; ++r)
        xout[(size_t)(m0 + r + 8 * half) * 64 + nt * 16 + l16] = acc[nt][r] + bias;
    }
  } else {
#pragma unroll
    for (int nt = 0; nt < 4; ++nt) {
      const float bias = bin[nt * 16 + l16];
#pragma unroll
      for (int r = 0; r < 8; ++r) {
        const int row = m0 + r + 8 * half;
        if (row < N) xout[(size_t)row * 64 + nt * 16 + l16] = acc[nt][r] + bias;
      }
    }
  }
}

// ---------------------------------------------------------------------------
// Kernel 2: edge aggregation.  agg[dst] += relu(x[src]) over all edges.
// Thread t -> edge t>>4, feature quad (t&15)*4.  float4 gathers hit L2.
// ---------------------------------------------------------------------------
__global__ __launch_bounds__(256)
void edge_agg_kernel(const int* __restrict__ src, const int* __restrict__ dst,
                     const float* __restrict__ x, float* __restrict__ agg,
                     unsigned total) {
  const unsigned t = blockIdx.x * 256u + threadIdx.x;
  if (t >= total) return;
  const int e  = (int)(t >> 4);
  const int f4 = (int)(t & 15u) * 4;
  const int s = src[e], d = dst[e];
  const float4 v = *reinterpret_cast<const float4*>(x + (size_t)s * 64 + f4);
  float* o = agg + (size_t)d * 64 + f4;
  atomicAdd(o + 0, fmaxf(v.x, 0.0f));
  atomicAdd(o + 1, fmaxf(v.y, 0.0f));
  atomicAdd(o + 2, fmaxf(v.z, 0.0f));
  atomicAdd(o + 3, fmaxf(v.w, 0.0f));
}

// ---------------------------------------------------------------------------
// Kernel 3: h[N,128] = ((1+eps)*x + agg) @ W1[64,128] + b1
// One wave -> 16-row strip (8 col tiles); combine fused into A-frag load,
// amortized over all 8 tiles.  16 K-steps of 4.
// ---------------------------------------------------------------------------
__global__ __launch_bounds__(256)
void gin_gemm1_kernel(const float* __restrict__ x, const float* __restrict__ agg,
                      const float* __restrict__ W1, const float* __restrict__ b1,
                      const float* __restrict__ epss, int layer,
                      float* __restrict__ h, int N, int mtiles) {
  const int lane  = threadIdx.x & 31;
  const int mtile = blockIdx.x * 8 + (threadIdx.x >> 5);
  if (mtile >= mtiles) return;
  const int m0 = mtile * 16;
  const int half = lane >> 4, l16 = lane & 15;
  const int mrow = min(m0 + l16, N - 1);
  const float es = 1.0f + epss[layer];   // uniform scalar load
  const size_t arow = (size_t)mrow * 64;
  v8f acc[8] = {};
#pragma unroll
  for (int kk = 0; kk < 64; kk += 4) {
    const int ka = kk + 2 * half;
    v2f a;
    a.x = es * x[arow + ka]     + agg[arow + ka];
    a.y = es * x[arow + ka + 1] + agg[arow + ka + 1];
#pragma unroll
    for (int nt = 0; nt < 8; ++nt) {
      v2f b;
      b.x = W1[(size_t)ka * 128 + nt * 16 + l16];
      b.y = W1[(size_t)(ka + 1) * 128 + nt * 16 + l16];
      acc[nt] = wmma_f32_k4(a, b, acc[nt]);
    }
  }
  if (m0 + 16 <= N) {
#pragma unroll
    for (int nt = 0; nt < 8; ++nt) {
      const float bias = b1[nt * 16 + l16];
#pragma unroll
      for (int r = 0; r < 8; ++r)
        h[(size_t)(m0 + r + 8 * half) * 128 + nt * 16 + l16] = acc[nt][r] + bias;
    }
  } else {
#pragma unroll
    for (int nt = 0; nt < 8; ++nt) {
      const float bias = b1[nt * 16 + l16];
#pragma unroll
      for (int r = 0; r < 8; ++r) {
        const int row = m0 + r + 8 * half;
        if (row < N) h[(size_t)row * 128 + nt * 16 + l16] = acc[nt][r] + bias;
      }
    }
  }
}

// ---------------------------------------------------------------------------
// Kernel 4a: per-channel sum / sumsq over N rows of h[N,128].
// ---------------------------------------------------------------------------
__global__ __launch_bounds__(256)
void bn_stats_kernel(const float* __restrict__ h, float* __restrict__ stats, int N) {
  const int tid = threadIdx.x;
  const int t   = blockIdx.x * 256 + tid;
  const int ch  = t & 127;
  const int rstride = (gridDim.x * 256) >> 7;
  float s = 0.0f, q = 0.0f;
  for (int row = t >> 7; row < N; row += rstride) {
    const float v = h[(size_t)row * 128 + ch];
    s += v;
    q += v * v;
  }
  __shared__ float sh_s[256], sh_q[256];
  sh_s[tid] = s; sh_q[tid] = q;
  __syncthreads();
  if (tid < 128) {
    atomicAdd(&stats[tid],       sh_s[tid] + sh_s[tid + 128]);
    atomicAdd(&stats[128 + tid], sh_q[tid] + sh_q[tid + 128]);
  }
}

// Kernel 4b: fold BN into per-channel scale/shift:  y = h*scale + shift
__global__ void bn_finalize_kernel(const float* __restrict__ stats,
                                   const float* __restrict__ gamma,
                                   const float* __restrict__ beta,
                                   float* __restrict__ ss, int N) {
  const int c = threadIdx.x;                 // 128 threads
  const float invN = 1.0f / (float)N;
  const float mu  = stats[c] * invN;
  const float var = stats[128 + c] * invN - mu * mu;
  const float sc  = gamma[c] * rsqrtf(var + 1e-5f);
  ss[c]       = sc;
  ss[128 + c] = beta[c] - mu * sc;
}

// ---------------------------------------------------------------------------
// Kernel 5: x[N,64] += relu(h*scale+shift) @ W2[128,64] + b2
// One wave -> 16-row strip (4 col tiles).  BN scale/shift staged in LDS;
// BN+ReLU fused into A-frag load; residual fused into the store.
// ---------------------------------------------------------------------------
__global__ __launch_bounds__(256)
void gin_gemm2_kernel(const float* __restrict__ h, const float* __restrict__ ss,
                      const float* __restrict__ W2, const float* __restrict__ b2,
                      float* __restrict__ x, int N, int mtiles) {
  __shared__ float s_ss[256];
  s_ss[threadIdx.x] = ss[threadIdx.x];       // 256 floats: scale[128]|shift[128]
  __syncthreads();

  const int lane  = threadIdx.x & 31;
  const int mtile = blockIdx.x * 8 + (threadIdx.x >> 5);
  if (mtile >= mtiles) return;
  const int m0 = mtile * 16;
  const int half = lane >> 4, l16 = lane & 15;
  const int mrow = min(m0 + l16, N - 1);
  const size_t arow = (size_t)mrow * 128;
  v8f acc[4] = {};
#pragma unroll
  for (int kk = 0; kk < 128; kk += 4) {
    const int ka = kk + 2 * half;
    v2f a;
    a.x = fmaxf(h[arow + ka]     * s_ss[ka]     + s_ss[128 + ka],     0.0f);
    a.y = fmaxf(h[arow + ka + 1] * s_ss[ka + 1] + s_ss[128 + ka + 1], 0.0f);
#pragma unroll
    for (int nt = 0; nt < 4; ++nt) {
      v2f b;
      b.x = W2[(size_t)ka * 64 + nt * 16 + l16];
      b.y = W2[(size_t)(ka + 1) * 64 + nt * 16 + l16];
      acc[nt] = wmma_f32_k4(a, b, acc[nt]);
    }
  }
  if (m0 + 16 <= N) {
#pragma unroll
    for (int nt = 0; nt < 4; ++nt) {
      const float bias = b2[nt * 16 + l16];
#pragma unroll
      for (int r = 0; r < 8; ++r) {
        const size_t idx = (size_t)(m0 + r + 8 * half) * 64 + nt * 16 + l16;
        x[idx] = x[idx] + acc[nt][r] + bias;   // residual
      }
    }
  } else {
#pragma unroll
    for (int nt = 0; nt < 4; ++nt) {
      const float bias = b2[nt * 16 + l16];
#pragma unroll
      for (int r = 0; r < 8; ++r) {
        const int row = m0 + r + 8 * half;
        if (row < N) {
          const size_t idx = (size_t)row * 64 + nt * 16 + l16;
          x[idx] = x[idx] + acc[nt][r] + bias;
        }
      }
    }
  }
}

// ---------------------------------------------------------------------------
// Kernel 6: pooled[G=64,64] += x by graph id (batch).
// ---------------------------------------------------------------------------
__global__ __launch_bounds__(256)
void pool_kernel(const float* __restrict__ x, const int* __restrict__ batch,
                 float* __restrict__ pooled, unsigned total) {
  const unsigned t = blockIdx.x * 256u + threadIdx.x;
  if (t >= total) return;
  const int n  = (int)(t >> 4);
  const int f4 = (int)(t & 15u) * 4;
  const int g = batch[n];
  const float4 v = *reinterpret_cast<const float4*>(x + (size_t)n * 64 + f4);
  float* o = pooled + (size_t)g * 64 + f4;
  atomicAdd(o + 0, v.x);
  atomicAdd(o + 1, v.y);
  atomicAdd(o + 2, v.z);
  atomicAdd(o + 3, v.w);
}

// Kernel 7: out[g] = pooled[g,:] . Wout + bout
__global__ void out_kernel(const float* __restrict__ pooled,
                           const float* __restrict__ Wout,
                           const float* __restrict__ bout,
                           float* __restrict__ out) {
  const int g = threadIdx.x;                 // 64 threads
  float acc = bout[0];
#pragma unroll
  for (int f = 0; f < 64; ++f) acc += pooled[(size_t)g * 64 + f] * Wout[f];
  out[g] = acc;
}

// ---------------------------------------------------------------------------
extern "C" void kernel_launch(void* const* d_in, const int* in_sizes, int n_in,
                              void* d_out, int out_size, void* d_ws, size_t ws_size,
                              hipStream_t stream) {
  // Inputs (setup_inputs order)
  const float* x_in   = (const float*)d_in[0];      // [N,32]
  const int*   eidx   = (const int*)  d_in[1];      // [2,E]
  const int*   batch  = (const int*)  d_in[2];      // [N]
  const float* Win    = (const float*)d_in[3];      // [32,64]
  const float* bin    = (const float*)d_in[4];      // [64]
  const float* W1s    = (const float*)d_in[5];      // [L,64,128]
  const float* b1s    = (const float*)d_in[6];      // [L,128]
  const float* gammas = (const float*)d_in[7];      // [L,128]
  const float* betas  = (const float*)d_in[8];      // [L,128]
  const float* W2s    = (const float*)d_in[9];      // [L,128,64]
  const float* b2s    = (const float*)d_in[10];     // [L,64]
  const float* epss   = (const float*)d_in[11];     // [L]
  const float* Wout   = (const float*)d_in[12];     // [64,1]
  const float* bout   = (const float*)d_in[13];     // [1]

  const int N = in_sizes[0] / 32;
  const int E = in_sizes[1] / 2;
  const int L = in_sizes[11];
  const int* src = eidx;
  const int* dst = eidx + E;

  // Workspace layout (all float): x[N*64] | agg[N*64] | h[N*128] | stats[256]
  //                               | scaleshift[256] | pooled[64*64]
  float* xbuf   = (float*)d_ws;
  float* agg    = xbuf  + (size_t)N * 64;
  float* h      = agg   + (size_t)N * 64;
  float* stats  = h     + (size_t)N * 128;
  float* ss     = stats + 256;
  float* pooled = ss    + 256;

  const int mtiles = (N + 15) / 16;
  const int mblocks = (mtiles + 7) / 8;      // 8 waves (strips) per block
  const dim3 blk(256);

  // Input projection
  gemm_in_kernel<<<mblocks, blk, 0, stream>>>(x_in, Win, bin, xbuf, N, mtiles);

  for (int l = 0; l < L; ++l) {
    hipMemsetAsync(agg, 0, (size_t)N * 64 * sizeof(float), stream);
    hipMemsetAsync(stats, 0, 256 * sizeof(float), stream);

    const unsigned etotal = (unsigned)E * 16u;
    edge_agg_kernel<<<(etotal + 255u) / 256u, blk, 0, stream>>>(src, dst, xbuf, agg, etotal);

    gin_gemm1_kernel<<<mblocks, blk, 0, stream>>>(
        xbuf, agg, W1s + (size_t)l * 64 * 128, b1s + (size_t)l * 128,
        epss, l, h, N, mtiles);

    bn_stats_kernel<<<256, blk, 0, stream>>>(h, stats, N);
    bn_finalize_kernel<<<1, 128, 0, stream>>>(stats, gammas + (size_t)l * 128,
                                              betas + (size_t)l * 128, ss, N);

    gin_gemm2_kernel<<<mblocks, blk, 0, stream>>>(
        h, ss, W2s + (size_t)l * 128 * 64, b2s + (size_t)l * 64, xbuf, N, mtiles);
  }

  hipMemsetAsync(pooled, 0, 64 * 64 * sizeof(float), stream);
  const unsigned ptotal = (unsigned)N * 16u;
  pool_kernel<<<(ptotal + 255u) / 256u, blk, 0, stream>>>(xbuf, batch, pooled, ptotal);
  out_kernel<<<1, 64, 0, stream>>>(pooled, Wout, bout, (float*)d_out);
}